// VectorQuantizer_61624190763226
// MI455X (gfx1250) — compile-verified
//
#include <hip/hip_runtime.h>
#include <math.h>

#if defined(__HIP_DEVICE_COMPILE__)
#if !__has_builtin(__builtin_amdgcn_wmma_f32_16x16x4_f32)
#error "device pass: missing __builtin_amdgcn_wmma_f32_16x16x4_f32 on this toolchain"
#endif
#define WMMA_F32_16X16X4(A, B, Cacc) \
  __builtin_amdgcn_wmma_f32_16x16x4_f32(false, (A), false, (B), (short)0, (Cacc), false, false)
#else
// host pass: never executed; keeps host sema away from amdgcn builtins
#define WMMA_F32_16X16X4(A, B, Cacc) (Cacc)
#endif

typedef float v2f __attribute__((ext_vector_type(2)));
typedef float v8f __attribute__((ext_vector_type(8)));

// Problem constants (fixed by the reference)
#define N_ROWS   65536   // B*H*W = 64*32*32
#define C_DIM    64
#define K_CODES  1024
#define HW       1024    // H*W
#define CHW      65536   // C*H*W
#define CHUNK    128     // codes staged in LDS per pass
#define LSTRIDE  68      // LDS dword stride per code row: bank = (4n + c) % 64, conflict-free

// ---------------------------------------------------------------- init: zero histogram
__global__ void vq_init(float* __restrict__ hist) {
  int i = blockIdx.x * 256 + threadIdx.x;
  if (i < K_CODES) hist[i] = 0.0f;
}

// ---------------------------------------------------------------- ||e_k||^2
__global__ void vq_enorm(const float* __restrict__ emb, float* __restrict__ enorm) {
  int k = blockIdx.x * 256 + threadIdx.x;
  if (k < K_CODES) {
    const float* e = emb + (size_t)k * C_DIM;
    float s = 0.0f;
#pragma unroll
    for (int c = 0; c < C_DIM; ++c) s = fmaf(e[c], e[c], s);
    enorm[k] = s;
  }
}

// ---------------------------------------------------------------- main VQ kernel
// grid 512 x block 256 (8 waves); each wave: 16 rows vs all 1024 codes via fp32 WMMA.
__global__ __launch_bounds__(256) void vq_main(
    const float* __restrict__ x,      // (64,64,32,32): x[b*CHW + c*HW + p]
    const float* __restrict__ emb,    // (1024,64) row-major
    const float* __restrict__ enorm,  // (1024,)
    float* __restrict__ out_q,        // N_ROWS * C_DIM
    float* __restrict__ out_idx,      // N_ROWS (indices as float)
    float* __restrict__ wave_sse,     // N_ROWS/16 per-wave SSE partials
    float* __restrict__ hist)         // K_CODES usage counts
{
  __shared__ float se[CHUNK * LSTRIDE];  // ~34.8 KB swizzled codebook chunk
  __shared__ float sen[CHUNK];           // chunk ||e||^2

  const int tid  = threadIdx.x;
  const int lane = tid & 31;
  const int wv   = tid >> 5;
  const int wid  = blockIdx.x * 8 + wv;
  const int row0 = wid << 4;             // 16 rows per wave

  const int lrow = lane & 15;            // row (A) / code (B) index within tile
  const int half = lane >> 4;            // 0 or 1
  const int par  = half << 1;            // K parity offset: 0 or 2

  // ---- A tile: lane holds x[row0+lrow][4j+par], x[row0+lrow][4j+par+1], j=0..15 ----
  // row0 is a multiple of 16 and 16 | HW, so all 16 rows share one image b.
  const int b_img = row0 >> 10;
  const int p0    = row0 & (HW - 1);
  const float* xb = x + (size_t)b_img * CHW + (size_t)(p0 + lrow);
  v2f a[16];
  float xn_part = 0.0f;
#pragma unroll
  for (int j = 0; j < 16; ++j) {
    float a0 = xb[(size_t)(4 * j + par) * HW];
    float a1 = xb[(size_t)(4 * j + par + 1) * HW];
    a[j].x = a0; a[j].y = a1;
    xn_part = fmaf(a0, a0, xn_part);
    xn_part = fmaf(a1, a1, xn_part);
  }
  // full ||x_row||^2: combine the two K-parity halves (lane l <-> l^16)
  float xn_row = xn_part + __shfl_xor(xn_part, 16, 32);
  // xnv[v] = ||x_{v + 8*half}||^2  (C/D layout: VGPR v -> row v + 8*half)
  float xnv[8];
#pragma unroll
  for (int v = 0; v < 8; ++v)
    xnv[v] = __shfl(xn_row, v + (half << 3), 32);

  float bestD[8];
  int   bestI[8];
#pragma unroll
  for (int v = 0; v < 8; ++v) { bestD[v] = 3.402823466e38f; bestI[v] = 0; }

  for (int kc = 0; kc < K_CODES; kc += CHUNK) {
    __syncthreads();
    // cooperative stage: 128 codes x 64 floats -> swizzled LDS (stride 68)
    {
      const float* g = emb + (size_t)kc * C_DIM;
#pragma unroll
      for (int it = 0; it < 8; ++it) {
        int e4 = it * 256 + tid;          // float4 slot, 2048 total
        int n  = e4 >> 4;                 // 16 float4 per code row
        int c4 = (e4 & 15) << 2;
        float4 vdat = *(const float4*)(g + (size_t)n * C_DIM + c4);
        *(float4*)(&se[n * LSTRIDE + c4]) = vdat;
      }
      if (tid < CHUNK) sen[tid] = enorm[kc + tid];
    }
    __syncthreads();

    for (int t = 0; t < CHUNK / 16; ++t) {
      const int nl = t * 16;
      const float en = sen[nl + lrow];                       // ||e||^2 for this lane's column
      const float* bb = &se[(nl + lrow) * LSTRIDE + par];
      v8f acc = {0.f, 0.f, 0.f, 0.f, 0.f, 0.f, 0.f, 0.f};
#pragma unroll
      for (int j = 0; j < 16; ++j) {
        v2f bj = *(const v2f*)(bb + 4 * j);                  // ds_load_b64, conflict-free
        acc = WMMA_F32_16X16X4(a[j], bj, acc);
      }
      const int idxc = kc + nl + lrow;
#pragma unroll
      for (int v = 0; v < 8; ++v) {
        float d = fmaf(-2.0f, acc[v], xnv[v] + en);          // ||x||^2 - 2s + ||e||^2
        if (d < bestD[v]) { bestD[v] = d; bestI[v] = idxc; } // strict < keeps lowest idx
      }
    }
  }

  // ---- argmin across the 16 lanes of each half (tie -> lower index) ----
#pragma unroll
  for (int v = 0; v < 8; ++v) {
#pragma unroll
    for (int off = 8; off >= 1; off >>= 1) {
      float dO = __shfl_xor(bestD[v], off, 32);
      int   iO = __shfl_xor(bestI[v], off, 32);
      if (dO < bestD[v] || (dO == bestD[v] && iO < bestI[v])) {
        bestD[v] = dO; bestI[v] = iO;
      }
    }
  }
  // now every lane of half h holds final (d, idx) for rows h*8 + v, v=0..7

  // per-wave SSE = sum of min distances over 16 rows (deterministic)
  float sseh = 0.0f;
#pragma unroll
  for (int v = 0; v < 8; ++v) sseh += bestD[v];
  float sse16 = sseh + __shfl_xor(sseh, 16, 32);
  if (lane == 0) wave_sse[wid] = sse16;

  // indices + histogram (lanes 0 and 16 each cover their 8 rows)
  if (lrow == 0) {
#pragma unroll
    for (int v = 0; v < 8; ++v) {
      int r = row0 + (half << 3) + v;
      out_idx[r] = (float)bestI[v];
      atomicAdd(&hist[bestI[v]], 1.0f);   // exact integer-valued float adds
    }
  }

  // quantized output: gather emb[idx] rows, fully coalesced (float2 per lane)
#pragma unroll
  for (int r = 0; r < 16; ++r) {
    int idxr = __shfl(bestI[r & 7], (r & 8) ? 16 : 0, 32);
    float2 q = *(const float2*)(emb + (size_t)idxr * C_DIM + (lane << 1));
    *(float2*)(out_q + (size_t)(row0 + r) * C_DIM + (lane << 1)) = q;
  }
}

// ---------------------------------------------------------------- finalize: loss + perplexity
__global__ __launch_bounds__(256) void vq_finalize(
    const float* __restrict__ wave_sse, const float* __restrict__ hist,
    float* __restrict__ out_loss, float* __restrict__ out_perp)
{
  __shared__ float rs[256], rh[256];
  int t = threadIdx.x;
  float s = 0.0f;
  for (int i = t; i < N_ROWS / 16; i += 256) s += wave_sse[i];
  float h = 0.0f;
  for (int i = t; i < K_CODES; i += 256) {
    float p = hist[i] * (1.0f / (float)N_ROWS);
    h += p * logf(p + 1e-10f);
  }
  rs[t] = s; rh[t] = h;
  __syncthreads();
  for (int o = 128; o > 0; o >>= 1) {
    if (t < o) { rs[t] += rs[t + o]; rh[t] += rh[t + o]; }
    __syncthreads();
  }
  if (t == 0) {
    // vq_loss = (1 + 0.25) * SSE / (N*C)
    *out_loss = 1.25f * rs[0] * (1.0f / (float)(N_ROWS * C_DIM));
    *out_perp = expf(-rh[0]);
  }
}

// ---------------------------------------------------------------- launch
extern "C" void kernel_launch(void* const* d_in, const int* in_sizes, int n_in,
                              void* d_out, int out_size, void* d_ws, size_t ws_size,
                              hipStream_t stream) {
  (void)in_sizes; (void)n_in; (void)out_size; (void)ws_size;

  const float* x   = (const float*)d_in[0];   // (64,64,32,32) f32
  const float* emb = (const float*)d_in[1];   // (1024,64) f32
  float* out = (float*)d_out;
  float* ws  = (float*)d_ws;

  // workspace layout (24 KB)
  float* wave_sse = ws;              // 4096
  float* hist     = ws + 4096;       // 1024
  float* enorm    = ws + 5120;       // 1024

  // output layout: [quantized (N*C)] [vq_loss] [indices (N)] [perplexity]
  float* out_q    = out;
  float* out_loss = out + (size_t)N_ROWS * C_DIM;
  float* out_idx  = out_loss + 1;
  float* out_perp = out_idx + N_ROWS;

  vq_init    <<<4,   256, 0, stream>>>(hist);
  vq_enorm   <<<4,   256, 0, stream>>>(emb, enorm);
  vq_main    <<<512, 256, 0, stream>>>(x, emb, enorm, out_q, out_idx, wave_sse, hist);
  vq_finalize<<<1,   256, 0, stream>>>(wave_sse, hist, out_loss, out_perp);
}